// block_model_20770461843916
// MI455X (gfx1250) — compile-verified
//
#include <hip/hip_runtime.h>

typedef __attribute__((ext_vector_type(2))) float v2f;
typedef __attribute__((ext_vector_type(8))) float v8f;
typedef __attribute__((ext_vector_type(4))) int   v4i;

#define B_DIM 128
#define C_DIM 512
#define LIN   720
#define LOUT  336
#define KC    16
#define NCH   (LIN / KC)   // 45 k-chunks
#define NTILE 48           // columns per block
#define XS_STRIDE 20       // X rows: 16 k + 4 pad floats (conflict-free A b64 reads)
#define WS_STRIDE 20       // W cols (n-major): 16 k + 4 pad floats (conflict-free B b64 reads)

#if defined(__gfx1250__) && \
    __has_builtin(__builtin_amdgcn_global_load_async_to_lds_b128) && \
    __has_builtin(__builtin_amdgcn_global_load_async_to_lds_b32)
#define USE_ASYNC 1
#else
#define USE_ASYNC 0
#endif

#if USE_ASYNC
#define AS1 __attribute__((address_space(1)))
#define AS3 __attribute__((address_space(3)))
__device__ __forceinline__ void async_b128(const float* g, float* l) {
    __builtin_amdgcn_global_load_async_to_lds_b128((AS1 v4i*)g, (AS3 v4i*)l, 0, 0);
}
__device__ __forceinline__ void async_b32(const float* g, float* l) {
    __builtin_amdgcn_global_load_async_to_lds_b32((AS1 int*)g, (AS3 int*)l, 0, 0);
}
__device__ __forceinline__ void wait_async() {
#if __has_builtin(__builtin_amdgcn_s_wait_asynccnt)
    __builtin_amdgcn_s_wait_asynccnt(0);
#else
    asm volatile("s_wait_asynccnt 0x0" ::: "memory");
#endif
}
#endif

__global__ __launch_bounds__(128)
void grouped_linear_wmma_f32(const float* __restrict__ x,
                             const float* __restrict__ w,
                             const float* __restrict__ bias,
                             float* __restrict__ out) {
    __shared__ float Xs[2][B_DIM * XS_STRIDE];  // 2 * 128*20*4 = 20480 B
    __shared__ float Ws[2][NTILE * WS_STRIDE];  // 2 *  48*20*4 =  7680 B, layout [n][k]

    const int tid  = threadIdx.x;
    const int wave = tid >> 5;        // 0..3 -> rows [wave*32, wave*32+32)
    const int lane = tid & 31;
    const int half = lane >> 4;       // K sub-select per f32-WMMA operand layout
    const int l16  = lane & 15;

    const int c     = blockIdx.y;
    const int nBase = blockIdx.x * NTILE;

    const float* xc = x + (size_t)c * LIN;               // + m*(C*LIN) + k
    const float* wc = w + (size_t)c * LIN * LOUT;        // + k*LOUT + n

    // X loader: 128 rows x 16 k = 512 float4, 4 per thread (f4 = tid + i*128)
    int xm[4], xk[4];
    #pragma unroll
    for (int i = 0; i < 4; ++i) {
        const int f4 = tid + i * 128;
        xm[i] = f4 >> 2;
        xk[i] = (f4 & 3) << 2;
    }

    v8f acc00 = {}, acc01 = {}, acc02 = {};
    v8f acc10 = {}, acc11 = {}, acc12 = {};

#if USE_ASYNC
    // W loader: 16 k x 48 n = 768 scalar copies, 6 per thread; transpose happens via
    // per-lane LDS destination addresses (global reads stay coalesced along n).
    int wk[6], wn[6];
    #pragma unroll
    for (int j = 0; j < 6; ++j) {
        const int f = tid + j * 128;
        wk[j] = f / 48;
        wn[j] = f % 48;
    }
    auto issue_chunk = [&](int kBase, int buf) {
        #pragma unroll
        for (int i = 0; i < 4; ++i)
            async_b128(xc + (size_t)xm[i] * ((size_t)C_DIM * LIN) + kBase + xk[i],
                       &Xs[buf][xm[i] * XS_STRIDE + xk[i]]);
        #pragma unroll
        for (int j = 0; j < 6; ++j)
            async_b32(wc + (size_t)(kBase + wk[j]) * LOUT + nBase + wn[j],
                      &Ws[buf][wn[j] * WS_STRIDE + wk[j]]);
    };

    issue_chunk(0, 0);
    wait_async();
    __syncthreads();
#else
    // Fallback: reg-staged pipeline (also what the host pass parses).
    const int wrow0 = tid / 12,         wcol0 = (tid % 12) << 2;
    const int wrow1 = (tid + 128) / 12, wcol1 = ((tid + 128) % 12) << 2;
    const bool doW1 = (tid < 64);
    float4 xr[4], wr0, wr1;
    auto load_regs = [&](int kBase) {
        #pragma unroll
        for (int i = 0; i < 4; ++i)
            xr[i] = *(const float4*)(xc + (size_t)xm[i] * ((size_t)C_DIM * LIN) + kBase + xk[i]);
        wr0 = *(const float4*)(wc + (size_t)(kBase + wrow0) * LOUT + nBase + wcol0);
        if (doW1)
            wr1 = *(const float4*)(wc + (size_t)(kBase + wrow1) * LOUT + nBase + wcol1);
    };
    auto store_lds = [&](int buf) {
        #pragma unroll
        for (int i = 0; i < 4; ++i)
            *(float4*)&Xs[buf][xm[i] * XS_STRIDE + xk[i]] = xr[i];
        float* wd = &Ws[buf][wcol0 * WS_STRIDE + wrow0];
        wd[0 * WS_STRIDE] = wr0.x; wd[1 * WS_STRIDE] = wr0.y;
        wd[2 * WS_STRIDE] = wr0.z; wd[3 * WS_STRIDE] = wr0.w;
        if (doW1) {
            float* we = &Ws[buf][wcol1 * WS_STRIDE + wrow1];
            we[0 * WS_STRIDE] = wr1.x; we[1 * WS_STRIDE] = wr1.y;
            we[2 * WS_STRIDE] = wr1.z; we[3 * WS_STRIDE] = wr1.w;
        }
    };
    load_regs(0);
    store_lds(0);
    __syncthreads();
#endif

    for (int ch = 0; ch < NCH; ++ch) {
        const int cur = ch & 1;
#if USE_ASYNC
        if (ch + 1 < NCH) issue_chunk((ch + 1) * KC, (ch + 1) & 1);  // DMA overlaps compute
#else
        if (ch + 1 < NCH) load_regs((ch + 1) * KC);
#endif

        const float* xs0 = &Xs[cur][(wave * 32 + l16) * XS_STRIDE + 2 * half];
        const float* xs1 = xs0 + 16 * XS_STRIDE;
        const float* ws  = &Ws[cur][l16 * WS_STRIDE + 2 * half];
        #pragma unroll
        for (int ks = 0; ks < KC; ks += 4) {
            v2f a0 = *(const v2f*)(xs0 + ks);                 // A m-tile 0 (ds_load_b64)
            v2f a1 = *(const v2f*)(xs1 + ks);                 // A m-tile 1
            v2f b0 = *(const v2f*)(ws + ks);                  // B n-tiles, n-major LDS
            v2f b1 = *(const v2f*)(ws + 16 * WS_STRIDE + ks);
            v2f b2 = *(const v2f*)(ws + 32 * WS_STRIDE + ks);
            acc00 = __builtin_amdgcn_wmma_f32_16x16x4_f32(false, a0, false, b0, (short)0, acc00, false, false);
            acc01 = __builtin_amdgcn_wmma_f32_16x16x4_f32(false, a0, false, b1, (short)0, acc01, false, false);
            acc02 = __builtin_amdgcn_wmma_f32_16x16x4_f32(false, a0, false, b2, (short)0, acc02, false, false);
            acc10 = __builtin_amdgcn_wmma_f32_16x16x4_f32(false, a1, false, b0, (short)0, acc10, false, false);
            acc11 = __builtin_amdgcn_wmma_f32_16x16x4_f32(false, a1, false, b1, (short)0, acc11, false, false);
            acc12 = __builtin_amdgcn_wmma_f32_16x16x4_f32(false, a1, false, b2, (short)0, acc12, false, false);
        }

#if USE_ASYNC
        wait_async();                 // our copies for chunk ch+1 have landed in LDS
#else
        if (ch + 1 < NCH) store_lds((ch + 1) & 1);
#endif
        __syncthreads();
    }

    // Epilogue: D element (vgpr j, lane) = row (j + 8*half), col l16. Add bias, store.
    const float* bc  = bias + (size_t)c * LOUT + nBase;
    const float bv0 = bc[ 0 + l16];
    const float bv1 = bc[16 + l16];
    const float bv2 = bc[32 + l16];

    #pragma unroll
    for (int j = 0; j < 8; ++j) {
        const int m0 = wave * 32 + j + 8 * half;        // batch index, m-tile 0
        float* o0 = out + ((size_t)m0 * C_DIM + c) * LOUT + nBase + l16;
        o0[ 0] = acc00[j] + bv0;
        o0[16] = acc01[j] + bv1;
        o0[32] = acc02[j] + bv2;
        float* o1 = o0 + (size_t)16 * C_DIM * LOUT;     // m-tile 1 (rows +16)
        o1[ 0] = acc10[j] + bv0;
        o1[16] = acc11[j] + bv1;
        o1[32] = acc12[j] + bv2;
    }
}

extern "C" void kernel_launch(void* const* d_in, const int* in_sizes, int n_in,
                              void* d_out, int out_size, void* d_ws, size_t ws_size,
                              hipStream_t stream) {
    (void)in_sizes; (void)n_in; (void)d_ws; (void)ws_size; (void)out_size;
    const float* x    = (const float*)d_in[0];
    const float* w    = (const float*)d_in[1];
    const float* bias = (const float*)d_in[2];
    float* out = (float*)d_out;

    dim3 grid(LOUT / NTILE, C_DIM);  // (7, 512)
    dim3 block(128);                 // 4 waves, each owns 32 rows x 48 cols
    grouped_linear_wmma_f32<<<grid, block, 0, stream>>>(x, w, bias, out);
}